// GCN_53755810677200
// MI455X (gfx1250) — compile-verified
//
#include <hip/hip_runtime.h>

typedef __attribute__((ext_vector_type(2))) float v2f;
typedef __attribute__((ext_vector_type(8))) float v8f;

// ---------------- small utility kernels ----------------

__global__ void set_ones_f32(float* __restrict__ p, int n) {
  int i = blockIdx.x * blockDim.x + threadIdx.x;
  if (i < n) p[i] = 1.0f;
}

__global__ void zero_f32(float* __restrict__ p, long n) {
  long i = (long)blockIdx.x * blockDim.x + threadIdx.x;
  if (i < n) p[i] = 0.0f;
}

// deg[dst] += 1 per edge (self-loop "1 +" handled by set_ones init)
__global__ void deg_count(const int* __restrict__ dst, float* __restrict__ deg, int e) {
  int i = blockIdx.x * blockDim.x + threadIdx.x;
  if (i < e) atomicAdd(&deg[dst[i]], 1.0f);
}

__global__ void rsqrt_inplace(float* __restrict__ p, int n) {
  int i = blockIdx.x * blockDim.x + threadIdx.x;
  if (i < n) p[i] = rsqrtf(p[i]);
}

// ---------------- WMMA f32 GEMM: H[nrows,64] = X[nrows,64] @ W[64,64] ----------------
// 256 threads = 8 waves; each wave computes a 16x64 strip via v_wmma_f32_16x16x4_f32.
// A layout (16x4 f32): lanes 0-15: {K=0,K=1}; lanes 16-31: {K=2,K=3} for row M=lane&15.
// B layout (4x16 f32): VGPR0 = row (K=koff), VGPR1 = row (K=koff+1), col N = lane&15.
// C/D (16x16 f32): VGPR r = row (lane<16 ? r : r+8), col N = lane&15.
__global__ __launch_bounds__(256) void gemm_n64_wmma(
    const float* __restrict__ X, const float* __restrict__ W,
    float* __restrict__ H, int nrows)
{
  __shared__ float lds_x[128 * 65];  // stride 65 breaks bank conflicts on row-strided reads
  __shared__ float lds_w[64 * 64];

  const int tid = threadIdx.x;
  const int row_base = blockIdx.x * 128;

  // coalesced staging
  for (int i = tid; i < 64 * 64; i += 256) lds_w[i] = W[i];
  for (int i = tid; i < 128 * 64; i += 256) {
    int r = i >> 6, c = i & 63;
    int gr = row_base + r;
    lds_x[r * 65 + c] = (gr < nrows) ? X[(long)gr * 64 + c] : 0.0f;
  }
  __syncthreads();

  const int wave = tid >> 5;
  const int lane = tid & 31;
  const int m    = lane & 15;
  const int hf   = lane >> 4;       // half-wave selector
  const int koff = hf * 2;          // A/B k sub-offset per ISA layout
  const int m0   = wave * 16;       // local row strip

  v8f acc0 = {}, acc1 = {}, acc2 = {}, acc3 = {};

#pragma unroll
  for (int kk = 0; kk < 64; kk += 4) {
    v2f a;
    a.x = lds_x[(m0 + m) * 65 + kk + koff];
    a.y = lds_x[(m0 + m) * 65 + kk + koff + 1];

    const int kr0 = (kk + koff) * 64 + m;
    const int kr1 = kr0 + 64;
    v2f b0, b1, b2, b3;
    b0.x = lds_w[kr0 +  0]; b0.y = lds_w[kr1 +  0];
    b1.x = lds_w[kr0 + 16]; b1.y = lds_w[kr1 + 16];
    b2.x = lds_w[kr0 + 32]; b2.y = lds_w[kr1 + 32];
    b3.x = lds_w[kr0 + 48]; b3.y = lds_w[kr1 + 48];

    acc0 = __builtin_amdgcn_wmma_f32_16x16x4_f32(false, a, false, b0, (short)0, acc0, false, false);
    acc1 = __builtin_amdgcn_wmma_f32_16x16x4_f32(false, a, false, b1, (short)0, acc1, false, false);
    acc2 = __builtin_amdgcn_wmma_f32_16x16x4_f32(false, a, false, b2, (short)0, acc2, false, false);
    acc3 = __builtin_amdgcn_wmma_f32_16x16x4_f32(false, a, false, b3, (short)0, acc3, false, false);
  }

#pragma unroll
  for (int r = 0; r < 8; ++r) {
    int grow = row_base + m0 + hf * 8 + r;
    if (grow < nrows) {
      long base = (long)grow * 64 + m;
      H[base +  0] = acc0[r];
      H[base + 16] = acc1[r];
      H[base + 32] = acc2[r];
      H[base + 48] = acc3[r];
    }
  }
}

// ---------------- edge gather/scale/scatter-add (roofline-dominant) ----------------
// 64 consecutive threads handle one edge's 64-float feature row: coalesced gather,
// f32 global atomic scatter (L2-resident RMW; agg fits in 192MB L2).
__global__ __launch_bounds__(256) void edge_scatter(
    const float* __restrict__ h, const int* __restrict__ src, const int* __restrict__ dst,
    const float* __restrict__ dinv, float* __restrict__ agg, int nedges)
{
  int e = blockIdx.x * 4 + (threadIdx.x >> 6);
  if (e >= nedges) return;
  int f = threadIdx.x & 63;
  int s = src[e], d = dst[e];
  float nrm = dinv[s] * dinv[d];
  atomicAdd(&agg[(long)d * 64 + f], h[(long)s * 64 + f] * nrm);
}

// out = relu(agg + h*dinv^2 + bias)   (elementwise; safe for out aliasing agg or h)
__global__ void combine_bias_relu(
    const float* __restrict__ agg, const float* __restrict__ h,
    const float* __restrict__ dinv, const float* __restrict__ bias,
    float* __restrict__ out, int n)
{
  long i = (long)blockIdx.x * blockDim.x + threadIdx.x;
  if (i >= (long)n * 64) return;
  int node = (int)(i >> 6);
  int f = (int)(i & 63);
  float di = dinv[node];
  float v = agg[i] + h[i] * (di * di) + bias[f];
  out[i] = fmaxf(v, 0.0f);
}

// ---------------- head: out[node] = h[node,:] . Wo + bo  (one wave32 per node) ----------------
__global__ __launch_bounds__(256) void output_proj(
    const float* __restrict__ h, const float* __restrict__ Wo,
    const float* __restrict__ bo, float* __restrict__ out, int n)
{
  int node = (blockIdx.x * 256 + threadIdx.x) >> 5;
  int lane = threadIdx.x & 31;
  if (node >= n) return;
  const float* row = h + (long)node * 64;
  float a = row[lane] * Wo[lane] + row[lane + 32] * Wo[lane + 32];
#pragma unroll
  for (int off = 16; off > 0; off >>= 1) a += __shfl_xor(a, off, 32);
  if (lane == 0) out[node] = a + bo[0];
}

// ---------------- driver ----------------

extern "C" void kernel_launch(void* const* d_in, const int* in_sizes, int n_in,
                              void* d_out, int out_size, void* d_ws, size_t ws_size,
                              hipStream_t stream) {
  const float* x  = (const float*)d_in[0];
  const int*   ei = (const int*)d_in[1];
  const float* W1 = (const float*)d_in[2];
  const float* b1 = (const float*)d_in[3];
  const float* W2 = (const float*)d_in[4];
  const float* b2 = (const float*)d_in[5];
  const float* Wo = (const float*)d_in[6];
  const float* bo = (const float*)d_in[7];
  float* out = (float*)d_out;

  const int n = in_sizes[0] / 64;   // 100000
  const int e = in_sizes[1] / 2;    // 1250000
  const int* src = ei;
  const int* dst = ei + e;

  // workspace: dinv [n] | bufA [n*64] | bufB [n*64]
  float* dinv = (float*)d_ws;
  float* bufA = dinv + n;
  float* bufB = bufA + (size_t)n * 64;

  const long nf = (long)n * 64;
  dim3 blk(256);
  int gN    = (n + 255) / 256;
  int gE    = (e + 255) / 256;
  int gNF   = (int)((nf + 255) / 256);
  int gE4   = (e + 3) / 4;
  int gGemm = (n + 127) / 128;
  int gProj = (n * 32 + 255) / 256;

  // deg -> dinv (deg = 1 + in-degree at dst)
  set_ones_f32 <<<gN, blk, 0, stream>>>(dinv, n);
  deg_count    <<<gE, blk, 0, stream>>>(dst, dinv, e);
  rsqrt_inplace<<<gN, blk, 0, stream>>>(dinv, n);

  // layer 1: h1 = x@W1 (bufA); agg (bufB); out1 = relu(...) -> bufA
  gemm_n64_wmma    <<<gGemm, blk, 0, stream>>>(x, W1, bufA, n);
  zero_f32         <<<gNF,   blk, 0, stream>>>(bufB, nf);
  edge_scatter     <<<gE4,   blk, 0, stream>>>(bufA, src, dst, dinv, bufB, e);
  combine_bias_relu<<<gNF,   blk, 0, stream>>>(bufB, bufA, dinv, b1, bufA, n);

  // layer 2: h2 = out1@W2 (bufB); agg (bufA); out2 = relu(...) -> bufB
  gemm_n64_wmma    <<<gGemm, blk, 0, stream>>>(bufA, W2, bufB, n);
  zero_f32         <<<gNF,   blk, 0, stream>>>(bufA, nf);
  edge_scatter     <<<gE4,   blk, 0, stream>>>(bufB, src, dst, dinv, bufA, e);
  combine_bias_relu<<<gNF,   blk, 0, stream>>>(bufA, bufB, dinv, b2, bufB, n);

  // head
  output_proj<<<gProj, blk, 0, stream>>>(bufB, Wo, bo, out, n);
}